// fre_loss_764504179117
// MI455X (gfx1250) — compile-verified
//
#include <hip/hip_runtime.h>
#include <hip/hip_bf16.h>
#include <math.h>

typedef __attribute__((ext_vector_type(2))) float v2f;
typedef __attribute__((ext_vector_type(8))) float v8f;

#define N_SAMPLES 60000
#define K_BINS    1400
#define TC        25            // time chunks
#define CHUNK     2400          // N_SAMPLES / TC
#define S_SUB     8             // subchunks per chunk (B columns = 8 x 2 signals)
#define L_SUB     300           // CHUNK / S_SUB
#define U_UNROLL  5             // unroll factor; 75 k-steps = 15 * 5
#define JITER     15            // (L_SUB/4) / U_UNROLL
#define BG        175           // K_BINS / 8 bin-groups (8 bins per wave)

#define AS3 __attribute__((address_space(3)))

// ---------------------------------------------------------------------------
// Kernel 1: sparse-DFT via WMMA f32 16x16x4, all 16 columns live.
//   A (16x4): rows 0-7 = cos(2*pi*f_b*(tbase+k)), rows 8-15 = sin (phase -1/4)
//   B (4x16): col 2s+sig = hann*sig(tbase + s*L_SUB + k)
//   C (16x16): rows 0-7 = R_s (cos partials), rows 8-15 = I_s (sin partials)
//   Recombine: re = sum_s cos(phi_s) R_s - sin(phi_s) I_s, phi_s = 2pi f s L
//              im = sum_s sin(phi_s) R_s + cos(phi_s) I_s
// One wave per block; grid = (BG, TC).
// ws: part[(chunk*K_BINS + bin)*4 + comp], comp: 0=re_p 1=im_p 2=re_l 3=im_l
// ---------------------------------------------------------------------------
__global__ void __launch_bounds__(32)
dft_wmma_kernel(const float* __restrict__ preds,
                const float* __restrict__ labels,
                const int*   __restrict__ FsPtr,
                const float* __restrict__ bpm,
                float*       __restrict__ part)
{
    __shared__ float lds_s[2 * CHUNK];      // [0,CHUNK)=preds, [CHUNK,2C)=labels
    __shared__ float cpart[16 * 16];        // C dump: [col][hi][j] = col*16+hi*8+j

    const int lane  = threadIdx.x;          // 0..31
    const int bg    = blockIdx.x;           // bin group
    const int c     = blockIdx.y;           // time chunk
    const int b0    = bg * 8;
    const int tbase = c * CHUNK;

    const int hi   = lane >> 4;             // 0: lanes 0-15, 1: lanes 16-31
    const int m    = lane & 15;             // A row / B-C column
    const int col  = m;
    const int koff = hi * 2;                // this lane's K pair {koff, koff+1}

    const int Fsv = *FsPtr;

    // --- async-stage raw signals into LDS (ASYNCcnt path) ------------------
    {
        const unsigned ldsBase =
            (unsigned)(unsigned long long)(AS3 float*)lds_s;  // raw LDS byte offset
        for (int i = lane; i < CHUNK; i += 32) {
            const unsigned            dp = ldsBase + (unsigned)(i) * 4u;
            const unsigned            dl = ldsBase + (unsigned)(CHUNK + i) * 4u;
            const unsigned long long  gp = (unsigned long long)(const void*)&preds[tbase + i];
            const unsigned long long  gl = (unsigned long long)(const void*)&labels[tbase + i];
            asm volatile("global_load_async_to_lds_b32 %0, %1, off"
                         :: "v"(dp), "v"(gp) : "memory");
            asm volatile("global_load_async_to_lds_b32 %0, %1, off"
                         :: "v"(dl), "v"(gl) : "memory");
        }
        asm volatile("s_wait_asynccnt 0x0" ::: "memory");
    }
    __syncthreads();

    // --- apply Hann window in place (one v_cos_f32 per element) ------------
    // np.hanning: w[t] = 0.5 - 0.5*cos(2*pi*t/(N-1)); hw cos takes revolutions
    const float invNm1 = 1.0f / (float)(N_SAMPLES - 1);
    for (int i = lane; i < CHUNK; i += 32) {
        const int   t = tbase + i;
        const float w = 0.5f - 0.5f * __builtin_amdgcn_cosf((float)t * invNm1);
        lds_s[i]         *= w;
        lds_s[CHUNK + i] *= w;
    }
    __syncthreads();

    // --- per-lane phase setup: fp64 init, fp32 increments (revolutions) ----
    // f = bpm/60/(Fs/N) cycles over N samples -> bpm/(60*Fs) rev/sample
    const int    bi    = b0 + (m & 7);
    const bool   isSin = (m >= 8);
    const double fd    = (double)bpm[bi] / (60.0 * (double)Fsv);

    double dstep = (double)(4 * U_UNROLL) * fd; dstep -= floor(dstep);
    const float drU = (float)dstep;         // phase step per unrolled iter

    float r0[U_UNROLL], r1[U_UNROLL];       // 10 independent trans chains
    #pragma unroll
    for (int u = 0; u < U_UNROLL; ++u) {
        double q0 = (double)(tbase + 4 * u + koff) * fd;     q0 -= floor(q0);
        double q1 = (double)(tbase + 4 * u + koff + 1) * fd; q1 -= floor(q1);
        if (isSin) {                         // sin(2pi r) = cos(2pi(r - 1/4))
            q0 -= 0.25; if (q0 < 0.0) q0 += 1.0;
            q1 -= 0.25; if (q1 < 0.0) q1 += 1.0;
        }
        r0[u] = (float)q0;
        r1[u] = (float)q1;
    }

    // --- B source for this lane's column: subchunk + signal ----------------
    const int sig  = col & 1;               // 0=preds, 1=labels
    const int ssub = col >> 1;              // subchunk 0..7
    const float* srcp = lds_s + sig * CHUNK + ssub * L_SUB + koff;

    v8f acc = {};
    for (int j = 0; j < JITER; ++j) {
        #pragma unroll
        for (int u = 0; u < U_UNROLL; ++u) {
            v2f a;
            a.x = __builtin_amdgcn_cosf(r0[u]);
            a.y = __builtin_amdgcn_cosf(r1[u]);
            const v2f b = *(const v2f*)(srcp + (U_UNROLL * j + u) * 4); // ds_load_b64
            acc = __builtin_amdgcn_wmma_f32_16x16x4_f32(
                      /*neg_a=*/false, a, /*neg_b=*/false, b,
                      /*c_mod=*/(short)0, acc, /*reuse_a=*/false, /*reuse_b=*/false);
            r0[u] += drU; r0[u] -= floorf(r0[u]);
            r1[u] += drU; r1[u] -= floorf(r1[u]);
        }
    }

    // --- dump C to LDS, rotate partials, write all 32 outputs --------------
    #pragma unroll
    for (int j = 0; j < 8; ++j) cpart[col * 16 + hi * 8 + j] = acc[j];
    __syncthreads();

    // lane w -> output (bin j, comp): comp 0=re_p 1=im_p 2=re_l 3=im_l
    const int    jb   = lane & 7;
    const int    comp = lane >> 3;
    const int    sg   = comp >> 1;          // which signal
    const int    im   = comp & 1;           // 0=re, 1=im
    const double fdj  = (double)bpm[b0 + jb] / (60.0 * (double)Fsv);

    float val = 0.f;
    #pragma unroll
    for (int s = 0; s < S_SUB; ++s) {
        double th = fdj * (double)(s * L_SUB); th -= floor(th);   // phi_s (rev)
        const float ct = __builtin_amdgcn_cosf((float)th);
        double ts = th - 0.25; if (ts < 0.0) ts += 1.0;
        const float st = __builtin_amdgcn_cosf((float)ts);        // sin(phi_s)
        const float R = cpart[(2 * s + sg) * 16 + jb];
        const float I = cpart[(2 * s + sg) * 16 + 8 + jb];
        val += im ? (st * R + ct * I) : (ct * R - st * I);
    }
    part[((size_t)c * K_BINS + (b0 + jb)) * 4 + comp] = val;
}

// ---------------------------------------------------------------------------
// Kernel 2: single-workgroup finalize.
//   ca = re^2+im^2 per bin; psd norm = ca/max(ca) (the /sum cancels);
//   out = 1/(||p1-p2|| + 1e-8)
// ---------------------------------------------------------------------------
__global__ void __launch_bounds__(256)
finalize_kernel(const float* __restrict__ part, float* __restrict__ out)
{
    __shared__ float ca1[K_BINS];
    __shared__ float ca2[K_BINS];
    __shared__ float red[256];

    const int tid = threadIdx.x;

    for (int bin = tid; bin < K_BINS; bin += 256) {
        float rp = 0.f, ip = 0.f, rl = 0.f, il = 0.f;
        for (int c = 0; c < TC; ++c) {
            const float4 q = *(const float4*)&part[((size_t)c * K_BINS + bin) * 4];
            rp += q.x; ip += q.y; rl += q.z; il += q.w;
        }
        ca1[bin] = rp * rp + ip * ip;
        ca2[bin] = rl * rl + il * il;
    }
    __syncthreads();

    float m1 = 0.f, m2 = 0.f;               // ca >= 0, so 0 is a safe identity
    for (int bin = tid; bin < K_BINS; bin += 256) {
        m1 = fmaxf(m1, ca1[bin]);
        m2 = fmaxf(m2, ca2[bin]);
    }
    red[tid] = m1; __syncthreads();
    for (int s = 128; s > 0; s >>= 1) {
        if (tid < s) red[tid] = fmaxf(red[tid], red[tid + s]);
        __syncthreads();
    }
    m1 = red[0]; __syncthreads();

    red[tid] = m2; __syncthreads();
    for (int s = 128; s > 0; s >>= 1) {
        if (tid < s) red[tid] = fmaxf(red[tid], red[tid + s]);
        __syncthreads();
    }
    m2 = red[0]; __syncthreads();

    const float inv1 = 1.0f / m1;
    const float inv2 = 1.0f / m2;
    float ssum = 0.f;
    for (int bin = tid; bin < K_BINS; bin += 256) {
        const float d = ca1[bin] * inv1 - ca2[bin] * inv2;
        ssum += d * d;
    }
    red[tid] = ssum; __syncthreads();
    for (int s = 128; s > 0; s >>= 1) {
        if (tid < s) red[tid] += red[tid + s];
        __syncthreads();
    }
    if (tid == 0) out[0] = 1.0f / (sqrtf(red[0]) + 1e-8f);
}

// ---------------------------------------------------------------------------
extern "C" void kernel_launch(void* const* d_in, const int* in_sizes, int n_in,
                              void* d_out, int out_size, void* d_ws, size_t ws_size,
                              hipStream_t stream) {
    const float* preds  = (const float*)d_in[0];
    const float* labels = (const float*)d_in[1];
    const int*   Fs     = (const int*)d_in[2];
    const float* bpm    = (const float*)d_in[3];
    float*       out    = (float*)d_out;
    float*       part   = (float*)d_ws;   // TC * K_BINS * 4 floats = 560 KB

    dim3 grid(BG, TC);
    dft_wmma_kernel<<<grid, 32, 0, stream>>>(preds, labels, Fs, bpm, part);
    finalize_kernel<<<1, 256, 0, stream>>>(part, out);
}